// EdgeGraphConv_69441031242030
// MI455X (gfx1250) — compile-verified
//
#include <hip/hip_runtime.h>

typedef __attribute__((ext_vector_type(2))) float v2f;
typedef __attribute__((ext_vector_type(8))) float v8f;

#define IN_FEATS 128
#define OUT_FEATS 64

// ---------------------------------------------------------------------------
// Kernel 0: zero the accumulator (d_out) and the degree array (in d_ws).
// ---------------------------------------------------------------------------
__global__ __launch_bounds__(256) void egc_zero_kernel(float* __restrict__ acc,
                                                       float* __restrict__ deg,
                                                       int accCount, int degCount) {
  int i = blockIdx.x * blockDim.x + threadIdx.x;
  if (i < accCount) acc[i] = 0.0f;
  if (i < degCount) deg[i] = 0.0f;
}

// ---------------------------------------------------------------------------
// Kernel 1: h = node_feat @ W_node + b_node via V_WMMA_F32_16X16X4_F32.
// Block = 256 threads = 8 waves. W_node staged in LDS in K-pair-packed
// layout: sWp[p*64 + n] = (W[2p][n], W[2p+1][n]) so each B fragment is a
// single aligned ds_load_b64 straight into an even VGPR pair (no shuffles).
// Each wave computes a 16(M) x 64(N) tile: 4 v8f accumulators,
// 32 K-steps x 4 WMMA ops.
// ---------------------------------------------------------------------------
__global__ __launch_bounds__(256) void egc_node_gemm_wmma(
    const float* __restrict__ A,     // [N,128]
    const float* __restrict__ W,     // [128,64]
    const float* __restrict__ bias,  // [64]
    float* __restrict__ H,           // [N,64] (workspace)
    int nNodes) {
  __shared__ v2f sWp[(IN_FEATS / 2) * OUT_FEATS];  // 32 KB
  __shared__ float sB[OUT_FEATS];

  const int tid = threadIdx.x;
  // Stage W as K-pairs: sWp[p*64+n] = {W[2p][n], W[2p+1][n]}
  for (int i = tid; i < (IN_FEATS / 2) * OUT_FEATS; i += 256) {
    const int p = i >> 6;        // pair index (kb/2)
    const int n = i & 63;        // column
    v2f t;
    t.x = W[(2 * p + 0) * OUT_FEATS + n];
    t.y = W[(2 * p + 1) * OUT_FEATS + n];
    sWp[i] = t;
  }
  if (tid < OUT_FEATS) sB[tid] = bias[tid];
  __syncthreads();

  const int wave = tid >> 5;
  const int lane = tid & 31;
  const int rowTile = blockIdx.x * 8 + wave;  // 16 rows per tile
  const int row0 = rowTile * 16;
  if (row0 >= nNodes) return;  // wave-uniform: EXEC stays all-ones for WMMA

  const int m = lane & 15;     // row within tile (for A), col (for B/C/D)
  const int half = lane >> 4;  // 0 or 1: selects K pair / row offset +8

  v8f acc0 = {}, acc1 = {}, acc2 = {}, acc3 = {};

  const float* arow = A + (size_t)(row0 + m) * IN_FEATS;

  for (int k = 0; k < IN_FEATS; k += 4) {
    // A 16x4 fragment: VGPR v, lane-half h -> A[m][k + v + 2h]
    v2f a = *(const v2f*)(arow + k + 2 * half);
    // B 4x16 fragments: VGPR v, lane-half h -> W[k + v + 2h][n]
    // pair index = (k + 2h)/2 = k/2 + h; one b64 LDS load per fragment.
    const v2f* wp = &sWp[(size_t)((k >> 1) + half) * OUT_FEATS + m];
    const v2f b0 = wp[0];
    const v2f b1 = wp[16];
    const v2f b2 = wp[32];
    const v2f b3 = wp[48];

    acc0 = __builtin_amdgcn_wmma_f32_16x16x4_f32(false, a, false, b0,
                                                 (short)0, acc0, false, false);
    acc1 = __builtin_amdgcn_wmma_f32_16x16x4_f32(false, a, false, b1,
                                                 (short)0, acc1, false, false);
    acc2 = __builtin_amdgcn_wmma_f32_16x16x4_f32(false, a, false, b2,
                                                 (short)0, acc2, false, false);
    acc3 = __builtin_amdgcn_wmma_f32_16x16x4_f32(false, a, false, b3,
                                                 (short)0, acc3, false, false);
  }

  // Epilogue: C/D layout -> lane (v, half) holds row (v + 8*half), col m.
  const float bb0 = sB[m + 0], bb1 = sB[m + 16], bb2 = sB[m + 32], bb3 = sB[m + 48];
#pragma unroll
  for (int v = 0; v < 8; ++v) {
    const int r = row0 + v + 8 * half;
    if (r < nNodes) {
      float* hr = H + (size_t)r * OUT_FEATS + m;
      hr[0]  = acc0[v] + bb0;
      hr[16] = acc1[v] + bb1;
      hr[32] = acc2[v] + bb2;
      hr[48] = acc3[v] + bb3;
    }
  }
}

// ---------------------------------------------------------------------------
// Kernel 2: per-edge fused edge-linear + gather + scatter-add.
// One wave per edge (grid-stride); lane handles 2 of 64 features (float2).
// Gather of h[src] is a fully-coalesced 256B read per wave (L2-resident:
// h is 25.6 MB << 192 MB L2). Scatter uses GLOBAL_ATOMIC_ADD_F32 into
// d_out; lane 0 bumps the degree counter once per edge.
// ---------------------------------------------------------------------------
__global__ __launch_bounds__(256) void egc_edge_scatter(
    const float* __restrict__ H,     // [N,64]
    const float* __restrict__ ef,    // [E,1]
    const float* __restrict__ We,    // [1,64]
    const float* __restrict__ be,    // [64]
    const int* __restrict__ src,
    const int* __restrict__ dst,
    float* __restrict__ outAcc,      // [N,64] accumulator (= d_out)
    float* __restrict__ deg,         // [N]
    int nEdges) {
  const int lane = threadIdx.x & 31;
  const int wave = (blockIdx.x * blockDim.x + threadIdx.x) >> 5;
  const int nWaves = (gridDim.x * blockDim.x) >> 5;
  const int j = lane * 2;

  const v2f w2 = *(const v2f*)(We + j);
  const v2f b2 = *(const v2f*)(be + j);

  for (int e = wave; e < nEdges; e += nWaves) {
    const int s = __builtin_amdgcn_readfirstlane(src[e]);
    const int d = __builtin_amdgcn_readfirstlane(dst[e]);
    const float f = ef[e];

    const v2f hh = *(const v2f*)(H + (size_t)s * OUT_FEATS + j);
    const float m0 = hh.x + f * w2.x + b2.x;
    const float m1 = hh.y + f * w2.y + b2.y;

    float* o = outAcc + (size_t)d * OUT_FEATS + j;
    atomicAdd(o + 0, m0);
    atomicAdd(o + 1, m1);
    if (lane == 0) atomicAdd(deg + d, 1.0f);
  }
}

// ---------------------------------------------------------------------------
// Kernel 3: out[i][:] /= max(deg[i], 1)
// ---------------------------------------------------------------------------
__global__ __launch_bounds__(256) void egc_finalize(float* __restrict__ outAcc,
                                                    const float* __restrict__ deg,
                                                    int total) {
  int i = blockIdx.x * blockDim.x + threadIdx.x;
  if (i < total) {
    const float dv = deg[i >> 6];  // i / 64
    outAcc[i] = outAcc[i] / fmaxf(dv, 1.0f);
  }
}

// ---------------------------------------------------------------------------
// Host launcher
// ---------------------------------------------------------------------------
extern "C" void kernel_launch(void* const* d_in, const int* in_sizes, int n_in,
                              void* d_out, int out_size, void* d_ws, size_t ws_size,
                              hipStream_t stream) {
  const float* node_feat = (const float*)d_in[0];  // [N,128]
  const float* edge_feat = (const float*)d_in[1];  // [E,1]
  const float* W_node    = (const float*)d_in[2];  // [128,64]
  const float* b_node    = (const float*)d_in[3];  // [64]
  const float* W_edge    = (const float*)d_in[4];  // [1,64]
  const float* b_edge    = (const float*)d_in[5];  // [64]
  const int*   src       = (const int*)d_in[6];    // [E]
  const int*   dst       = (const int*)d_in[7];    // [E]

  const int nNodes = in_sizes[0] / IN_FEATS;
  const int nEdges = in_sizes[6];

  float* outAcc = (float*)d_out;                   // [N,64], accumulate in place
  float* h      = (float*)d_ws;                    // [N,64] scratch
  float* deg    = h + (size_t)nNodes * OUT_FEATS;  // [N]

  // 0) zero accumulator + degrees
  {
    const int total = nNodes * OUT_FEATS;
    const int blocks = (total + 255) / 256;
    egc_zero_kernel<<<blocks, 256, 0, stream>>>(outAcc, deg, total, nNodes);
  }

  // 1) node GEMM (WMMA): h = node_feat @ W_node + b_node
  {
    const int rowTiles = (nNodes + 15) / 16;
    const int blocks = (rowTiles + 7) / 8;  // 8 waves (tiles) per block
    egc_node_gemm_wmma<<<blocks, 256, 0, stream>>>(node_feat, W_node, b_node, h, nNodes);
  }

  // 2) edge pass: gather + edge-linear + scatter-add
  {
    const int blocks = 8192;  // 65536 waves, grid-stride over 1.6M edges
    egc_edge_scatter<<<blocks, 256, 0, stream>>>(h, edge_feat, W_edge, b_edge,
                                                 src, dst, outAcc, deg, nEdges);
  }

  // 3) finalize: mean
  {
    const int total = nNodes * OUT_FEATS;
    const int blocks = (total + 255) / 256;
    egc_finalize<<<blocks, 256, 0, stream>>>(outAcc, deg, total);
  }
}